// Model_83519934038707
// MI455X (gfx1250) — compile-verified
//
#include <hip/hip_runtime.h>
#include <math.h>

typedef __attribute__((ext_vector_type(2))) float v2f;
typedef __attribute__((ext_vector_type(8))) float v8f;

#define N_NODES 4096
#define IN_DIM  512
#define H_DIM   128
#define OUT_DIM 40
#define KAPPA   0.8f
#define MAX_ITER 300
#define TOL     3e-6f

// epilogue flags
#define F_BIASMAT 1
#define F_BIASVEC 2
#define F_RELU    4
#define F_DIFF    8

// ---------------------------------------------------------------------------
// fp32 WMMA GEMM, fully specialized at compile time so the inner loop lowers
// to loads-with-immediate-offsets + v_wmma_f32_16x16x4_f32.
// C[M,N] = A[M,K] * B   where  B element (k,n) = B[k*LDB_ROW + n*LDB_COL].
// Each wave computes a 32(M) x 16(N) tile; launched waves == (M/32)*(N/16).
// ---------------------------------------------------------------------------
template<int M, int N, int KDIM, int LDA, int LDB_ROW, int LDB_COL, int LDC,
         int FLAGS, bool HAS_CONV>
__global__ void gemm_wmma(const float* __restrict__ A,
                          const float* __restrict__ B,
                          float* __restrict__ C,
                          const float* __restrict__ biasMat,
                          const float* __restrict__ biasVec,
                          const float* __restrict__ Xold,
                          unsigned int* __restrict__ errOut,
                          const int* __restrict__ conv)
{
    if (HAS_CONV && *conv) return;           // wave-uniform early exit
    int wave = (blockIdx.x * blockDim.x + threadIdx.x) >> 5;
    int lane = threadIdx.x & 31;
    constexpr int mPairs = M >> 5;
    int mp = wave % mPairs;
    int nt = wave / mPairs;
    int m0 = mp << 5;
    int n0 = nt << 4;

    int lm = lane & 15;                      // column / row-in-tile index
    int kh = (lane >> 4) << 1;               // 0 for lanes 0-15, 2 for 16-31

    v8f acc0 = {};
    v8f acc1 = {};

    const float* arow0 = A + (size_t)(m0 + lm) * LDA + kh;
    const float* arow1 = arow0 + (size_t)16 * LDA;
    const float* bbase = B + (size_t)kh * LDB_ROW + (size_t)(n0 + lm) * LDB_COL;

    for (int k = 0; k < KDIM; k += 4) {
        // A fragment: lane L holds A[m0+L, k+kh .. k+kh+1] (contiguous -> b64)
        v2f a0 = *(const v2f*)(arow0 + k);
        v2f a1 = *(const v2f*)(arow1 + k);
        // B fragment: v0 = row k+kh, v1 = row k+kh+1, column n0+lm
        const float* bp = bbase + (size_t)k * LDB_ROW;
        v2f b;
        b.x = bp[0];
        b.y = bp[LDB_ROW];
        acc0 = __builtin_amdgcn_wmma_f32_16x16x4_f32(false, a0, false, b,
                                                     (short)0, acc0, false, false);
        acc1 = __builtin_amdgcn_wmma_f32_16x16x4_f32(false, a1, false, b,
                                                     (short)0, acc1, false, false);
    }

    // Epilogue.  C layout: VGPR r holds rows m0+r (lanes 0-15) / m0+8+r (lanes 16-31)
    int mhi = (lane >> 4) << 3;
    int nIdx = n0 + lm;
    float bv = (FLAGS & F_BIASVEC) ? biasVec[nIdx] : 0.0f;
    float wmax = 0.0f;

#pragma unroll
    for (int r = 0; r < 8; ++r) {
        {
            int mrow = m0 + mhi + r;
            size_t idx = (size_t)mrow * LDC + nIdx;
            float v = acc0[r] + bv;
            if (FLAGS & F_BIASMAT) v += biasMat[idx];
            if (FLAGS & F_RELU)    v = fmaxf(v, 0.0f);
            if (FLAGS & F_DIFF)    wmax = fmaxf(wmax, fabsf(v - Xold[idx]));
            C[idx] = v;
        }
        {
            int mrow = m0 + 16 + mhi + r;
            size_t idx = (size_t)mrow * LDC + nIdx;
            float v = acc1[r] + bv;
            if (FLAGS & F_BIASMAT) v += biasMat[idx];
            if (FLAGS & F_RELU)    v = fmaxf(v, 0.0f);
            if (FLAGS & F_DIFF)    wmax = fmaxf(wmax, fabsf(v - Xold[idx]));
            C[idx] = v;
        }
    }
    if (FLAGS & F_DIFF)
        atomicMax(errOut, __float_as_uint(wmax));   // all values >= 0 -> bit order ok
}

// ---------------------------------------------------------------------------
// LayerNorm + exact GELU, also writes transpose U[h, n] = act
// ---------------------------------------------------------------------------
__global__ void ln_gelu_kernel(const float* __restrict__ hpre,
                               const float* __restrict__ g,
                               const float* __restrict__ b,
                               float* __restrict__ hact,
                               float* __restrict__ U)
{
    __shared__ float red[H_DIM];
    int row = blockIdx.x, t = threadIdx.x;
    float v = hpre[(size_t)row * H_DIM + t];
    red[t] = v; __syncthreads();
    for (int s = H_DIM / 2; s > 0; s >>= 1) { if (t < s) red[t] += red[t + s]; __syncthreads(); }
    float mu = red[0] * (1.0f / H_DIM);
    __syncthreads();
    float d = v - mu;
    red[t] = d * d; __syncthreads();
    for (int s = H_DIM / 2; s > 0; s >>= 1) { if (t < s) red[t] += red[t + s]; __syncthreads(); }
    float var = red[0] * (1.0f / H_DIM);
    float y = d * rsqrtf(var + 1e-5f) * g[t] + b[t];
    float act = 0.5f * y * (1.0f + erff(y * 0.70710678118654752f));
    hact[(size_t)row * H_DIM + t] = act;
    U[(size_t)t * N_NODES + row] = act;
}

// ---------------------------------------------------------------------------
// Residual + LayerNorm: Z[row,t] = LN(hact[row,t] + X[t, row])
// ---------------------------------------------------------------------------
__global__ void resln_kernel(const float* __restrict__ hact,
                             const float* __restrict__ X,
                             const float* __restrict__ g,
                             const float* __restrict__ b,
                             float* __restrict__ Z)
{
    __shared__ float red[H_DIM];
    int row = blockIdx.x, t = threadIdx.x;
    float v = hact[(size_t)row * H_DIM + t] + X[(size_t)t * N_NODES + row];
    red[t] = v; __syncthreads();
    for (int s = H_DIM / 2; s > 0; s >>= 1) { if (t < s) red[t] += red[t + s]; __syncthreads(); }
    float mu = red[0] * (1.0f / H_DIM);
    __syncthreads();
    float d = v - mu;
    red[t] = d * d; __syncthreads();
    for (int s = H_DIM / 2; s > 0; s >>= 1) { if (t < s) red[t] += red[t + s]; __syncthreads(); }
    float var = red[0] * (1.0f / H_DIM);
    Z[(size_t)row * H_DIM + t] = d * rsqrtf(var + 1e-5f) * g[t] + b[t];
}

// ---------------------------------------------------------------------------
// Row-wise projection of W (128x128) onto L1 ball of radius kappa.
// ---------------------------------------------------------------------------
__global__ void proj_kernel(const float* __restrict__ W, float* __restrict__ Wp)
{
    __shared__ float s[H_DIM];
    __shared__ float d[H_DIM];
    __shared__ float red[H_DIM];
    int row = blockIdx.x, t = threadIdx.x;
    float w = W[(size_t)row * H_DIM + t];
    float a = fabsf(w);

    red[t] = a; __syncthreads();
    for (int st = H_DIM / 2; st > 0; st >>= 1) { if (t < st) red[t] += red[t + st]; __syncthreads(); }
    float total = red[0];
    __syncthreads();
    if (total <= KAPPA) { Wp[(size_t)row * H_DIM + t] = w; return; }   // uniform per block

    s[t] = a; __syncthreads();
    for (int ksz = 2; ksz <= H_DIM; ksz <<= 1) {            // bitonic ascending
        for (int j = ksz >> 1; j > 0; j >>= 1) {
            int ixj = t ^ j;
            if (ixj > t) {
                bool up = ((t & ksz) == 0);
                float x = s[t], y = s[ixj];
                if ((up && x > y) || (!up && x < y)) { s[t] = y; s[ixj] = x; }
            }
            __syncthreads();
        }
    }
    d[t] = s[H_DIM - 1 - t];                                 // descending + scan
    __syncthreads();
    for (int off = 1; off < H_DIM; off <<= 1) {
        float add = (t >= off) ? d[t - off] : 0.0f;
        __syncthreads();
        d[t] += add;
        __syncthreads();
    }
    float sd  = s[H_DIM - 1 - t];
    float css = d[t];
    red[t] = (sd * (float)(t + 1) > css - KAPPA) ? 1.0f : 0.0f;
    __syncthreads();
    for (int st = H_DIM / 2; st > 0; st >>= 1) { if (t < st) red[t] += red[t + st]; __syncthreads(); }
    int rho = (int)(red[0] + 0.5f);
    float theta = (d[rho - 1] - KAPPA) / (float)rho;
    float pv = fmaxf(a - theta, 0.0f);
    Wp[(size_t)row * H_DIM + t] = (w > 0.0f) ? pv : ((w < 0.0f) ? -pv : 0.0f);
}

// ---------------------------------------------------------------------------
// Tiny decoder: out[row, o] = Z[row,:] . W_V[o,:] + b_V[o]
// ---------------------------------------------------------------------------
__global__ void decoder_kernel(const float* __restrict__ Z,
                               const float* __restrict__ WV,
                               const float* __restrict__ bV,
                               float* __restrict__ out)
{
    int idx = blockIdx.x * blockDim.x + threadIdx.x;
    if (idx >= N_NODES * OUT_DIM) return;
    int row = idx / OUT_DIM, o = idx % OUT_DIM;
    const float* z = Z + (size_t)row * H_DIM;
    const float* w = WV + (size_t)o * H_DIM;
    float acc = bV[o];
#pragma unroll 8
    for (int k = 0; k < H_DIM; ++k) acc = fmaf(z[k], w[k], acc);
    out[idx] = acc;
}

// ---------------------------------------------------------------------------
// state kernels
// ---------------------------------------------------------------------------
__global__ void init_kernel(float* __restrict__ X, int n,
                            int* __restrict__ conv, unsigned int* __restrict__ err)
{
    int i = blockIdx.x * blockDim.x + threadIdx.x;
    if (i < n) X[i] = 0.0f;
    if (i == 0) { *conv = 0; *err = 0u; }
}

__global__ void commit_kernel(const int* __restrict__ conv,
                              const float* __restrict__ Y, float* __restrict__ X, int n)
{
    if (*conv) return;
    int i = blockIdx.x * blockDim.x + threadIdx.x;
    if (i < n) X[i] = Y[i];
}

__global__ void check_kernel(int* __restrict__ conv, unsigned int* __restrict__ err)
{
    if (!*conv && __uint_as_float(*err) < TOL) *conv = 1;
    *err = 0u;
}

// ---------------------------------------------------------------------------
extern "C" void kernel_launch(void* const* d_in, const int* in_sizes, int n_in,
                              void* d_out, int out_size, void* d_ws, size_t ws_size,
                              hipStream_t stream)
{
    const float* x     = (const float*)d_in[0];    // [4096,512]
    const float* adj   = (const float*)d_in[3];    // [4096,4096]
    const float* W_enc = (const float*)d_in[4];    // [128,512]
    const float* b_enc = (const float*)d_in[5];    // [128]
    const float* ln_g  = (const float*)d_in[6];    // [128]
    const float* ln_b  = (const float*)d_in[7];    // [128]
    const float* Wmat  = (const float*)d_in[8];    // [128,128]
    const float* Om1   = (const float*)d_in[9];    // [128,128]
    const float* W_V   = (const float*)d_in[10];   // [40,128]
    const float* b_V   = (const float*)d_in[11];   // [40]
    float* out = (float*)d_out;

    // workspace carve-up (floats)
    const size_t NH = (size_t)N_NODES * H_DIM;     // 524288
    float* ws   = (float*)d_ws;
    float* hpre = ws;            ws += NH;
    float* hact = ws;            ws += NH;
    float* U    = ws;            ws += NH;
    float* Tbuf = ws;            ws += NH;
    float* Bmat = ws;            ws += NH;
    float* X    = ws;            ws += NH;
    float* Y    = ws;            ws += NH;
    float* Z    = ws;            ws += NH;
    float* Wp   = ws;            ws += (size_t)H_DIM * H_DIM;
    int* conv          = (int*)ws;
    unsigned int* err  = (unsigned int*)(ws + 1);

    const int WAVES  = (H_DIM >> 5) * (N_NODES >> 4);       // 1024 (also 4096/32 * 128/16)
    const int BLOCKS = WAVES / 4;                           // 128-thread blocks

    // 0) zero X and flags
    init_kernel<<<(int)((NH + 255) / 256), 256, 0, stream>>>(X, (int)NH, conv, err);

    // 1) encoder: hpre = x @ W_enc.T + b_enc   (B(k,n) = W_enc[n*512 + k])
    gemm_wmma<N_NODES, H_DIM, IN_DIM, IN_DIM, 1, IN_DIM, H_DIM,
              F_BIASVEC, false><<<BLOCKS, 128, 0, stream>>>(
        x, W_enc, hpre, nullptr, b_enc, nullptr, nullptr, nullptr);

    // 2) LN + GELU + transpose into U
    ln_gelu_kernel<<<N_NODES, H_DIM, 0, stream>>>(hpre, ln_g, ln_b, hact, U);
    // 3) project rows of W
    proj_kernel<<<H_DIM, H_DIM, 0, stream>>>(Wmat, Wp);

    // 4) Tbuf = Omega1 @ U   (M=128,K=128,N=4096)
    gemm_wmma<H_DIM, N_NODES, H_DIM, H_DIM, N_NODES, 1, N_NODES,
              0, false><<<BLOCKS, 128, 0, stream>>>(
        Om1, U, Tbuf, nullptr, nullptr, nullptr, nullptr, nullptr);

    // 5) Bmat = Tbuf @ adj   (M=128,K=4096,N=4096)
    gemm_wmma<H_DIM, N_NODES, N_NODES, N_NODES, N_NODES, 1, N_NODES,
              0, false><<<BLOCKS, 128, 0, stream>>>(
        Tbuf, adj, Bmat, nullptr, nullptr, nullptr, nullptr, nullptr);

    // 6) Picard iteration with device-side early exit
    for (int it = 0; it < MAX_ITER; ++it) {
        gemm_wmma<H_DIM, N_NODES, H_DIM, H_DIM, N_NODES, 1, N_NODES,
                  0, true><<<BLOCKS, 128, 0, stream>>>(
            Wp, X, Tbuf, nullptr, nullptr, nullptr, nullptr, conv);
        gemm_wmma<H_DIM, N_NODES, N_NODES, N_NODES, N_NODES, 1, N_NODES,
                  F_BIASMAT | F_RELU | F_DIFF, true><<<BLOCKS, 128, 0, stream>>>(
            Tbuf, adj, Y, Bmat, nullptr, X, err, conv);
        commit_kernel<<<(int)((NH + 255) / 256), 256, 0, stream>>>(conv, Y, X, (int)NH);
        check_kernel<<<1, 1, 0, stream>>>(conv, err);
    }

    // 7) residual + LN
    resln_kernel<<<N_NODES, H_DIM, 0, stream>>>(hact, X, ln_g, ln_b, Z);
    // 8) decoder
    decoder_kernel<<<(N_NODES * OUT_DIM + 255) / 256, 256, 0, stream>>>(Z, W_V, b_V, out);
}